// SelectiveSSM_75488345195322
// MI455X (gfx1250) — compile-verified
//
#include <hip/hip_runtime.h>
#include <hip/hip_bf16.h>
#include <math.h>

// Problem constants (from reference setup_inputs)
#define B_   2
#define L_   2048
#define D_   2048
#define R_   64            // DT_RANK
#define N_   16            // N_STATE
#define E_   96            // R_ + 2*N_
#define M_   (B_ * L_)     // 4096 rows of the token-major GEMMs
#define CH_  16            // scan chunks per sequence
#define CLEN_ (L_ / CH_)   // 128 steps per chunk

typedef float v2f __attribute__((ext_vector_type(2)));
typedef float v4f __attribute__((ext_vector_type(4)));
typedef float v8f __attribute__((ext_vector_type(8)));

// -----------------------------------------------------------------------------
// Kernel 1: x_dbl[M,96] = x[M,2048] * W_xproj[96,2048]^T   (FP32 WMMA 16x16x4)
// Block = 192 threads = 6 waves; wave w owns N-tile w (96 = 6*16 columns).
// A fragment: lane holds x[m0 + lane%16][kb + (lane/16)*2 + {0,1}]  -> float2
// B fragment: lane holds W[n0 + lane%16][kb + (lane/16)*2 + {0,1}]  -> float2
// -----------------------------------------------------------------------------
__global__ void __launch_bounds__(192)
xproj_gemm(const float* __restrict__ x, const float* __restrict__ Wx,
           float* __restrict__ xdbl) {
  const int lane = threadIdx.x & 31;
  const int wave = threadIdx.x >> 5;        // 0..5 -> N tile
  const int m0   = blockIdx.x * 16;         // M tile
  const int n0   = wave * 16;
  const int r    = lane & 15;               // A-row / B-col within tile
  const int koff = (lane >> 4) << 1;        // K sub-offset: 0 or 2
  const int half = lane >> 4;

  const float* xa = x  + (size_t)(m0 + r) * D_ + koff;
  const float* wb = Wx + (size_t)(n0 + r) * D_ + koff;

  v8f acc = {};
  #pragma unroll 8
  for (int kb = 0; kb < D_; kb += 4) {
    v2f a = *(const v2f*)(xa + kb);
    v2f b = *(const v2f*)(wb + kb);
    acc = __builtin_amdgcn_wmma_f32_16x16x4_f32(false, a, false, b,
                                                (short)0, acc, false, false);
  }
  // C/D layout: vgpr v, lane -> row m0 + v + 8*(lane/16), col n0 + lane%16
  #pragma unroll
  for (int v = 0; v < 8; ++v) {
    int m = m0 + v + half * 8;
    xdbl[(size_t)m * E_ + n0 + r] = acc[v];
  }
}

// -----------------------------------------------------------------------------
// Kernel 2: dt[M,2048] = softplus( dt_raw[M,64] * W_dt[2048,64]^T + b_dt )
// dt_raw = first 64 columns of x_dbl (row stride E_=96). K=64 -> 16 WMMAs/tile.
// Block = 256 threads = 8 waves, one 16x16 tile per wave.
// -----------------------------------------------------------------------------
__global__ void __launch_bounds__(256)
dt_gemm_softplus(const float* __restrict__ xdbl, const float* __restrict__ Wdt,
                 const float* __restrict__ bdt, float* __restrict__ dt) {
  const int lane = threadIdx.x & 31;
  const int wave = threadIdx.x >> 5;
  const int tile = blockIdx.x * 8 + wave;   // 0 .. 256*128-1
  const int mt   = tile / (D_ / 16);
  const int nt   = tile % (D_ / 16);
  const int m0   = mt * 16;
  const int d0   = nt * 16;
  const int r    = lane & 15;
  const int koff = (lane >> 4) << 1;
  const int half = lane >> 4;

  const float* xa = xdbl + (size_t)(m0 + r) * E_ + koff;  // dt_raw rows
  const float* wb = Wdt  + (size_t)(d0 + r) * R_ + koff;

  v8f acc = {};
  #pragma unroll
  for (int kb = 0; kb < R_; kb += 4) {
    v2f a = *(const v2f*)(xa + kb);
    v2f b = *(const v2f*)(wb + kb);
    acc = __builtin_amdgcn_wmma_f32_16x16x4_f32(false, a, false, b,
                                                (short)0, acc, false, false);
  }
  const int d   = d0 + r;
  const float bb = bdt[d];
  #pragma unroll
  for (int v = 0; v < 8; ++v) {
    int m   = m0 + v + half * 8;
    float z = acc[v] + bb;
    float sp = (z > 20.0f) ? z : log1pf(expf(z));  // stable softplus
    dt[(size_t)m * D_ + d] = sp;
  }
}

// -----------------------------------------------------------------------------
// Scan pass 1: per (b, chunk, d) compute local scan from h=0 over CLEN_ steps,
// recording chunk carry:  ap[n] = prod_t a_t[n],  h[n] = local end state.
// Lane mapping: d fastest (coalesced x/dt); B_ssm rows broadcast via float4.
// -----------------------------------------------------------------------------
__global__ void __launch_bounds__(256)
scan_pass1(const float* __restrict__ x, const float* __restrict__ xdbl,
           const float* __restrict__ dt, const float* __restrict__ A_log,
           float* __restrict__ carryA, float* __restrict__ carryH) {
  const int cid = blockIdx.x * 256 + threadIdx.x;  // 0 .. B_*CH_*D_-1
  const int d = cid % D_;
  const int t = cid / D_;
  const int c = t % CH_;
  const int b = t / CH_;

  float A[N_];
  #pragma unroll
  for (int n = 0; n < N_; ++n) A[n] = -expf(A_log[d * N_ + n]);

  float h[N_], ap[N_];
  #pragma unroll
  for (int n = 0; n < N_; ++n) { h[n] = 0.0f; ap[n] = 1.0f; }

  const int l0 = c * CLEN_;
  for (int ls = 0; ls < CLEN_; ++ls) {
    const size_t row = (size_t)(b * L_ + l0 + ls);
    const float xv  = x [row * D_ + d];
    const float dtv = dt[row * D_ + d];
    __builtin_prefetch(x  + (row + 4) * D_ + d, 0, 1);
    __builtin_prefetch(dt + (row + 4) * D_ + d, 0, 1);
    const float* bc = xdbl + row * E_ + R_;        // B_ssm row (16 floats)
    v4f Bq[4];
    #pragma unroll
    for (int q = 0; q < 4; ++q) Bq[q] = *(const v4f*)(bc + 4 * q);
    #pragma unroll
    for (int n = 0; n < N_; ++n) {
      float abar = expf(dtv * A[n]);
      float bn   = Bq[n >> 2][n & 3];
      h[n]  = abar * h[n] + dtv * bn * xv;
      ap[n] = ap[n] * abar;
    }
  }

  const size_t base = ((size_t)(b * CH_ + c) * N_) * D_ + d;
  #pragma unroll
  for (int n = 0; n < N_; ++n) {
    carryA[base + (size_t)n * D_] = ap[n];
    carryH[base + (size_t)n * D_] = h[n];
  }
}

// -----------------------------------------------------------------------------
// Scan pass 2: serial combine of the CH_ chunk carries per (b,d) channel.
// Rewrites carryH[c] with the chunk's START state (in-place, saves 4 MB ws).
// -----------------------------------------------------------------------------
__global__ void __launch_bounds__(256)
scan_pass2(const float* __restrict__ carryA, float* __restrict__ carryH) {
  const int cid = blockIdx.x * 256 + threadIdx.x;  // 0 .. B_*D_-1
  const int d = cid % D_;
  const int b = cid / D_;

  float h[N_];
  #pragma unroll
  for (int n = 0; n < N_; ++n) h[n] = 0.0f;

  for (int c = 0; c < CH_; ++c) {
    const size_t base = ((size_t)(b * CH_ + c) * N_) * D_ + d;
    #pragma unroll
    for (int n = 0; n < N_; ++n) {
      float a  = carryA[base + (size_t)n * D_];
      float he = carryH[base + (size_t)n * D_];
      carryH[base + (size_t)n * D_] = h[n];        // start state for chunk c
      h[n] = a * h[n] + he;
    }
  }
}

// -----------------------------------------------------------------------------
// Scan pass 3: re-run local scan seeded with chunk start state, emit
// y = sum_n h[n]*C[n] + x*D_param.
// -----------------------------------------------------------------------------
__global__ void __launch_bounds__(256)
scan_pass3(const float* __restrict__ x, const float* __restrict__ xdbl,
           const float* __restrict__ dt, const float* __restrict__ A_log,
           const float* __restrict__ Dp, const float* __restrict__ carryH,
           float* __restrict__ y) {
  const int cid = blockIdx.x * 256 + threadIdx.x;
  const int d = cid % D_;
  const int t = cid / D_;
  const int c = t % CH_;
  const int b = t / CH_;

  float A[N_];
  #pragma unroll
  for (int n = 0; n < N_; ++n) A[n] = -expf(A_log[d * N_ + n]);

  float h[N_];
  const size_t cbase = ((size_t)(b * CH_ + c) * N_) * D_ + d;
  #pragma unroll
  for (int n = 0; n < N_; ++n) h[n] = carryH[cbase + (size_t)n * D_];

  const float Dv = Dp[d];
  const int l0 = c * CLEN_;
  for (int ls = 0; ls < CLEN_; ++ls) {
    const size_t row = (size_t)(b * L_ + l0 + ls);
    const float xv  = x [row * D_ + d];
    const float dtv = dt[row * D_ + d];
    __builtin_prefetch(x  + (row + 4) * D_ + d, 0, 1);
    __builtin_prefetch(dt + (row + 4) * D_ + d, 0, 1);
    const float* bc = xdbl + row * E_ + R_;
    v4f Bq[4], Cq[4];
    #pragma unroll
    for (int q = 0; q < 4; ++q) {
      Bq[q] = *(const v4f*)(bc + 4 * q);
      Cq[q] = *(const v4f*)(bc + N_ + 4 * q);
    }
    float acc = 0.0f;
    #pragma unroll
    for (int n = 0; n < N_; ++n) {
      float abar = expf(dtv * A[n]);
      float bn   = Bq[n >> 2][n & 3];
      h[n] = abar * h[n] + dtv * bn * xv;
      acc += h[n] * Cq[n >> 2][n & 3];
    }
    y[row * D_ + d] = acc + xv * Dv;
  }
}

// -----------------------------------------------------------------------------
// Launch: inputs (x, W_xproj, W_dt, b_dt, A_log, D_param) all fp32.
// Workspace layout (floats): xdbl[M_*E_] | dt[M_*D_] | carryA | carryH
//   = 1.5 MB + 32 MB + 4 MB + 4 MB ~= 41.5 MB, fully overwritten each call.
// -----------------------------------------------------------------------------
extern "C" void kernel_launch(void* const* d_in, const int* in_sizes, int n_in,
                              void* d_out, int out_size, void* d_ws, size_t ws_size,
                              hipStream_t stream) {
  const float* x    = (const float*)d_in[0];
  const float* Wx   = (const float*)d_in[1];
  const float* Wdt  = (const float*)d_in[2];
  const float* bdt  = (const float*)d_in[3];
  const float* Alog = (const float*)d_in[4];
  const float* Dp   = (const float*)d_in[5];
  float* y = (float*)d_out;

  float* ws     = (float*)d_ws;
  float* xdbl   = ws;                                   // M_*E_
  float* dtbuf  = xdbl   + (size_t)M_ * E_;             // M_*D_
  float* carryA = dtbuf  + (size_t)M_ * D_;             // B_*CH_*N_*D_
  float* carryH = carryA + (size_t)B_ * CH_ * N_ * D_;  // B_*CH_*N_*D_

  xproj_gemm      <<<M_ / 16,                    192, 0, stream>>>(x, Wx, xdbl);
  dt_gemm_softplus<<<(M_ / 16) * (D_ / 16) / 8,  256, 0, stream>>>(xdbl, Wdt, bdt, dtbuf);
  scan_pass1      <<<(B_ * CH_ * D_) / 256,      256, 0, stream>>>(x, xdbl, dtbuf, Alog, carryA, carryH);
  scan_pass2      <<<(B_ * D_) / 256,            256, 0, stream>>>(carryA, carryH);
  scan_pass3      <<<(B_ * CH_ * D_) / 256,      256, 0, stream>>>(x, xdbl, dtbuf, Alog, Dp, carryH, y);
}